// ALIGNNEncoder_53687091200004
// MI455X (gfx1250) — compile-verified
//
#include <hip/hip_runtime.h>

// ============================================================================
// ALIGNN encoder forward for MI455X (gfx1250, wave32, WMMA).
// All GEMMs use V_WMMA_F32_16X16X4_F32 (native fp32 matrix pipe) since the
// reference is fp32 and K dims are small (4..129); the workload is L2-resident
// gather + skinny-GEMM bound, so fp32 WMMA + L2-cached gathers is the right
// operating point (192MB L2 holds every intermediate).
// ============================================================================

typedef __attribute__((ext_vector_type(2))) float v2f;
typedef __attribute__((ext_vector_type(8))) float v8f;

#define BN_EPS 1e-5f
#define WT_STRIDE 132   // 64-col LDS transpose stride (pad to avoid bank conflicts)

// ---------------------------------------------------------------------------
// Generic gathered dual-source WMMA GEMM:
//   Y[M,64] = gat(A0)[:,0:K1] @ W[0:K1,:] + gat(A1)[:,0:K2] @ W[K1:K1+K2,:]
//             (+ ea[row] * W[K1+K2,:] if ea)  + bias
// gat(A)[r] = A[g ? g[r] : r].  Block = 256 threads = 8 wave32; each wave owns
// a 16-row x 64-col output tile (4 WMMA accumulators).
// ---------------------------------------------------------------------------
__global__ __launch_bounds__(256) void k_gemm_wmma(
    const float* __restrict__ A0, int lda0, const int* __restrict__ g0,
    const float* __restrict__ A1, int lda1, const int* __restrict__ g1,
    const float* __restrict__ W, const float* __restrict__ bias,
    const float* __restrict__ ea,
    float* __restrict__ Y, int M, int K1, int K2)
{
  const int Ktot = K1 + K2;
  __shared__ float Wt[64 * WT_STRIDE];  // Wt[n][k] = W[k][n]
  __shared__ float Bs[64];
  __shared__ float We[64];

  for (int i = threadIdx.x; i < Ktot * 64; i += 256) {
    int k = i >> 6, n = i & 63;
    Wt[n * WT_STRIDE + k] = W[i];
  }
  if (threadIdx.x < 64) {
    Bs[threadIdx.x] = bias ? bias[threadIdx.x] : 0.0f;
    We[threadIdx.x] = ea ? W[(size_t)Ktot * 64 + threadIdx.x] : 0.0f;
  }
  __syncthreads();

  const int wave = threadIdx.x >> 5;
  const int lane = threadIdx.x & 31;
  const int half = lane >> 4;     // K-half for A/B fragments
  const int ln   = lane & 15;     // row (A) / col (B,C) within tile
  const int tileRow = (blockIdx.x * 8 + wave) * 16;
  if (tileRow >= M) return;       // wave-uniform exit; survivors have full EXEC

  int r = tileRow + ln;
  if (r >= M) r = M - 1;          // clamp (stores are guarded)
  const float* __restrict__ arow0 = A0 + (size_t)(g0 ? g0[r] : r) * lda0;
  const float* __restrict__ arow1 =
      (K2 > 0) ? (A1 + (size_t)(g1 ? g1[r] : r) * lda1) : A0;

  v8f acc[4] = {v8f{}, v8f{}, v8f{}, v8f{}};

  for (int k = 0; k < Ktot; k += 4) {
    const int kk = k + 2 * half;  // fp32 A layout: VGPR v holds K = 2*half + v
    v2f a;
    if (kk < K1) { a.x = arow0[kk];      a.y = arow0[kk + 1]; }
    else         { a.x = arow1[kk - K1]; a.y = arow1[kk - K1 + 1]; }
#pragma unroll
    for (int nt = 0; nt < 4; ++nt) {
      const float* wp = &Wt[(nt * 16 + ln) * WT_STRIDE + kk];
      v2f b; b.x = wp[0]; b.y = wp[1];   // B: lane=N, VGPR v holds K = 2*half+v
      acc[nt] = __builtin_amdgcn_wmma_f32_16x16x4_f32(
          false, a, false, b, (short)0, acc[nt], false, false);
    }
  }

  // C/D layout: VGPR vr -> row = tileRow + vr + 8*half, col = nt*16 + ln
  float eav[8];
#pragma unroll
  for (int vr = 0; vr < 8; ++vr) {
    int row = tileRow + vr + 8 * half;
    eav[vr] = (ea && row < M) ? ea[row] : 0.0f;
  }
#pragma unroll
  for (int nt = 0; nt < 4; ++nt) {
    int col = nt * 16 + ln;
    float bb = Bs[col];
    float wc = We[col];
#pragma unroll
    for (int vr = 0; vr < 8; ++vr) {
      int row = tileRow + vr + 8 * half;
      if (row < M)
        Y[(size_t)row * 64 + col] = acc[nt][vr] + bb + eav[vr] * wc;
    }
  }
}

// ---------------------------------------------------------------------------
// Column statistics: stats[0:64] += sum, stats[64:128] += sum of squares.
// Block = 256 = 4 row-groups x 64 cols, grid-stride over rows.
// ---------------------------------------------------------------------------
__global__ __launch_bounds__(256) void k_colstats(
    const float* __restrict__ X, int M, float* __restrict__ stats)
{
  const int col = threadIdx.x & 63;
  const int rg  = threadIdx.x >> 6;
  float s = 0.0f, sq = 0.0f;
  for (int r = blockIdx.x * 4 + rg; r < M; r += gridDim.x * 4) {
    float v = X[(size_t)r * 64 + col];
    s += v; sq += v * v;
  }
  __shared__ float shs[256], shq[256];
  shs[threadIdx.x] = s; shq[threadIdx.x] = sq;
  __syncthreads();
  if (rg == 0) {
    s  = shs[col] + shs[64 + col] + shs[128 + col] + shs[192 + col];
    sq = shq[col] + shq[64 + col] + shq[128 + col] + shq[192 + col];
    atomicAdd(&stats[col], s);
    atomicAdd(&stats[64 + col], sq);
  }
}

// BatchNorm (training stats, biased var) + SiLU (+ optional residual).
__global__ void k_bn_silu(const float* __restrict__ X, float* __restrict__ Y,
                          const float* __restrict__ stats,
                          const float* __restrict__ gamma,
                          const float* __restrict__ beta,
                          float invM, const float* __restrict__ residual,
                          size_t total)
{
  size_t i = (size_t)blockIdx.x * blockDim.x + threadIdx.x;
  if (i >= total) return;
  int c = (int)(i & 63);
  float mu  = stats[c] * invM;
  float var = stats[64 + c] * invM - mu * mu;
  float t = (X[i] - mu) * rsqrtf(var + BN_EPS) * gamma[c] + beta[c];
  float y = t / (1.0f + __expf(-t));
  if (residual) y += residual[i];
  Y[i] = y;
}

__global__ void k_zero(float* __restrict__ p, size_t n) {
  size_t i = (size_t)blockIdx.x * blockDim.x + threadIdx.x;
  if (i < n) p[i] = 0.0f;
}

// cnt[outer ? outer[idx[i]] : idx[i]] += 1
__global__ void k_count(const int* __restrict__ idx, const int* __restrict__ outer,
                        int M, float* __restrict__ cnt) {
  int i = blockIdx.x * blockDim.x + threadIdx.x;
  if (i >= M) return;
  int s = idx[i];
  if (outer) s = outer[s];
  atomicAdd(&cnt[s], 1.0f);
}

// out[seg(row)*ostride + coloff + col] += X[row*64 + col]
__global__ void k_seg_sum(const float* __restrict__ X, const int* __restrict__ idx,
                          const int* __restrict__ outer, int M,
                          float* __restrict__ out, int ostride, int coloff) {
  size_t i = (size_t)blockIdx.x * blockDim.x + threadIdx.x;
  size_t total = (size_t)M * 64;
  if (i >= total) return;
  int row = (int)(i >> 6), col = (int)(i & 63);
  int s = idx[row];
  if (outer) s = outer[s];
  atomicAdd(&out[(size_t)s * ostride + coloff + col], X[i]);
}

// X[r][c] /= max(cnt[r], 1)
__global__ void k_div_rows(float* __restrict__ X, const float* __restrict__ cnt,
                           size_t rows) {
  size_t i = (size_t)blockIdx.x * blockDim.x + threadIdx.x;
  if (i >= rows * 64) return;
  X[i] /= fmaxf(cnt[i >> 6], 1.0f);
}

// pooled[g][c] /= max(c<64 ? bc[g] : lbc[g], 1)   (pooled is [32,128])
__global__ void k_pool_div(float* __restrict__ pooled, const float* __restrict__ bc,
                           const float* __restrict__ lbc) {
  int i = blockIdx.x * blockDim.x + threadIdx.x;
  if (i >= 32 * 128) return;
  int g = i >> 7, c = i & 127;
  float cnt = (c < 64) ? bc[g] : lbc[g];
  pooled[i] /= fmaxf(cnt, 1.0f);
}

__global__ void k_silu(float* __restrict__ X, size_t n) {
  size_t i = (size_t)blockIdx.x * blockDim.x + threadIdx.x;
  if (i >= n) return;
  float t = X[i];
  X[i] = t / (1.0f + __expf(-t));
}

// ===========================================================================
// Host orchestration
// ===========================================================================

// Param leaf indices: JAX pytree flattening sorts dict keys alphabetically.
// Top level: atom_emb, atom_layers, b2a, line_emb, line_layers, out
// Conv leaves: bn1_b,bn1_g,bn2_b,bn2_g,msg_b1,msg_b2,msg_w1,msg_w2,upd_b,upd_w
// b2a leaves:  b, bn_b, bn_g, w
enum {
  IN_X = 0, IN_EI, IN_EA, IN_BATCH, IN_LX, IN_LEI, IN_LEA, IN_LMAP,
  P_AEMB_B = 8, P_AEMB_W = 9,
  P_ALAYER0 = 10,                 // 5 layers x 10 leaves -> 10..59
  P_B2A0 = 60,                    // 3 x 4 leaves        -> 60..71
  P_LEMB_B = 72, P_LEMB_W = 73,
  P_LLAYER0 = 74,                 // 3 layers x 10 leaves -> 74..103
  P_OUT_B = 104, P_OUT_W = 105
};
enum { CV_BN1_B = 0, CV_BN1_G, CV_BN2_B, CV_BN2_G, CV_MSG_B1, CV_MSG_B2,
       CV_MSG_W1, CV_MSG_W2, CV_UPD_B, CV_UPD_W };
enum { BA_B = 0, BA_BN_B, BA_BN_G, BA_W };

extern "C" void kernel_launch(void* const* d_in, const int* in_sizes, int n_in,
                              void* d_out, int out_size, void* d_ws, size_t ws_size,
                              hipStream_t stream)
{
  (void)n_in; (void)out_size; (void)ws_size;
  const int N  = in_sizes[0] / 92;   // atoms
  const int E  = in_sizes[2];        // edges (edge_attr is [E,1])
  const int LE = in_sizes[6];        // line-graph edges
  const int G  = 32;

  const float* x_raw  = (const float*)d_in[IN_X];
  const int*   ei     = (const int*)d_in[IN_EI];
  const float* eattr  = (const float*)d_in[IN_EA];
  const int*   batch  = (const int*)d_in[IN_BATCH];
  const float* lx_raw = (const float*)d_in[IN_LX];
  const int*   lei    = (const int*)d_in[IN_LEI];
  const float* leattr = (const float*)d_in[IN_LEA];
  const int*   lmap   = (const int*)d_in[IN_LMAP];

  const int* srcA = ei;       const int* dstA = ei + E;
  const int* srcL = lei;      const int* dstL = lei + LE;

  auto P = [&](int i) -> const float* { return (const float*)d_in[i]; };

  // ---- workspace carve-up (fp32) ----
  float* w = (float*)d_ws;
  float* xA    = w; w += (size_t)N * 64;
  float* xB    = w; w += (size_t)N * 64;
  float* lA    = w; w += (size_t)E * 64;
  float* lB    = w; w += (size_t)E * 64;
  float* msg   = w; w += (size_t)LE * 64;   // message buffer (largest)
  float* agg   = w; w += (size_t)E * 64;    // aggregation (N or E rows)
  float* cntAD = w; w += N;                 // in-degree (dst) atom graph
  float* cntLD = w; w += E;                 // in-degree (dst) line graph
  float* bond  = w; w += N;                 // src (bond) counts
  float* bc    = w; w += G;
  float* lbc   = w; w += G;
  float* stats = w; w += 128;
  float* pooled= w; w += (size_t)G * 128;

  // ---- launch helpers ----
  auto gemm = [&](float* Y, const float* A0, int lda0, const int* g0,
                  const float* A1, int lda1, const int* g1,
                  const float* W, const float* bias, const float* ea,
                  int M, int K1, int K2) {
    int grid = (M + 127) / 128;
    hipLaunchKernelGGL(k_gemm_wmma, dim3(grid), dim3(256), 0, stream,
                       A0, lda0, g0, A1, lda1, g1, W, bias, ea, Y, M, K1, K2);
  };
  auto zero = [&](float* p, size_t n) {
    hipLaunchKernelGGL(k_zero, dim3((unsigned)((n + 255) / 256)), dim3(256), 0,
                       stream, p, n);
  };
  auto colstats = [&](const float* X, int M) {
    zero(stats, 128);
    int grid = (M + 3) / 4; if (grid > 4096) grid = 4096; if (grid < 1) grid = 1;
    hipLaunchKernelGGL(k_colstats, dim3(grid), dim3(256), 0, stream, X, M, stats);
  };
  auto bn_silu = [&](const float* X, float* Y, const float* g, const float* b,
                     float invM, const float* res, size_t total) {
    hipLaunchKernelGGL(k_bn_silu, dim3((unsigned)((total + 255) / 256)), dim3(256),
                       0, stream, X, Y, stats, g, b, invM, res, total);
  };
  auto count = [&](const int* idx, const int* outer, int M, float* cnt) {
    hipLaunchKernelGGL(k_count, dim3((M + 255) / 256), dim3(256), 0, stream,
                       idx, outer, M, cnt);
  };
  auto seg_sum = [&](const float* X, const int* idx, const int* outer, int M,
                     float* out, int ostride, int coloff) {
    size_t total = (size_t)M * 64;
    hipLaunchKernelGGL(k_seg_sum, dim3((unsigned)((total + 255) / 256)), dim3(256),
                       0, stream, X, idx, outer, M, out, ostride, coloff);
  };
  auto div_rows = [&](float* X, const float* cnt, size_t rows) {
    hipLaunchKernelGGL(k_div_rows, dim3((unsigned)((rows * 64 + 255) / 256)),
                       dim3(256), 0, stream, X, cnt, rows);
  };

  // ---- one message-passing conv layer ----
  auto conv = [&](const float* Xin, float* Xout, int Nn,
                  const int* src, const int* dst, const float* ea_, int Ee,
                  const float* cnt, int pb) {
    // msg = silu(bn1(concat(x[dst], x[src], ea) @ W1 + b1))
    gemm(msg, Xin, 64, dst, Xin, 64, src, P(pb + CV_MSG_W1), P(pb + CV_MSG_B1),
         ea_, Ee, 64, 64);
    colstats(msg, Ee);
    bn_silu(msg, msg, P(pb + CV_BN1_G), P(pb + CV_BN1_B), 1.0f / Ee, nullptr,
            (size_t)Ee * 64);
    // msg = msg @ W2 + b2   (in-place: each 16-row tile reads only its own rows)
    gemm(msg, msg, 64, nullptr, nullptr, 0, nullptr, P(pb + CV_MSG_W2),
         P(pb + CV_MSG_B2), nullptr, Ee, 64, 0);
    // agg = segment_mean(msg, dst)
    zero(agg, (size_t)Nn * 64);
    seg_sum(msg, dst, nullptr, Ee, agg, 64, 0);
    div_rows(agg, cnt, Nn);
    // out = silu(bn2(concat(x, agg) @ updW + updB)) + x
    gemm(Xout, Xin, 64, nullptr, agg, 64, nullptr, P(pb + CV_UPD_W),
         P(pb + CV_UPD_B), nullptr, Nn, 64, 64);
    colstats(Xout, Nn);
    bn_silu(Xout, Xout, P(pb + CV_BN2_G), P(pb + CV_BN2_B), 1.0f / Nn, Xin,
            (size_t)Nn * 64);
  };

  // ---- degree counts (constant across layers) ----
  zero(cntAD, N);  count(dstA, nullptr, E, cntAD);
  zero(cntLD, E);  count(dstL, nullptr, LE, cntLD);
  zero(bond, N);   count(srcA, nullptr, E, bond);

  // ---- embeddings ----
  gemm(xA, x_raw, 92, nullptr, nullptr, 0, nullptr, P(P_AEMB_W), P(P_AEMB_B),
       nullptr, N, 92, 0);
  gemm(lA, lx_raw, 4, nullptr, nullptr, 0, nullptr, P(P_LEMB_W), P(P_LEMB_B),
       nullptr, E, 4, 0);

  float* x_cur = xA; float* x_nxt = xB;
  float* l_cur = lA; float* l_nxt = lB;

  // ---- 3 joint layers: atom conv, line conv, b2a fusion ----
  for (int i = 0; i < 3; ++i) {
    conv(x_cur, x_nxt, N, srcA, dstA, eattr, E, cntAD, P_ALAYER0 + i * 10);
    { float* t = x_cur; x_cur = x_nxt; x_nxt = t; }
    conv(l_cur, l_nxt, E, srcL, dstL, leattr, LE, cntLD, P_LLAYER0 + i * 10);
    { float* t = l_cur; l_cur = l_nxt; l_nxt = t; }
    // abf = segment_sum(line_x over src atoms) / bond_count
    zero(agg, (size_t)N * 64);
    seg_sum(l_cur, srcA, nullptr, E, agg, 64, 0);
    div_rows(agg, bond, N);
    // x = silu(bn(concat(x, abf) @ W + b))
    int pb = P_B2A0 + i * 4;
    gemm(x_nxt, x_cur, 64, nullptr, agg, 64, nullptr, P(pb + BA_W), P(pb + BA_B),
         nullptr, N, 64, 64);
    colstats(x_nxt, N);
    bn_silu(x_nxt, x_nxt, P(pb + BA_BN_G), P(pb + BA_BN_B), 1.0f / N, nullptr,
            (size_t)N * 64);
    { float* t = x_cur; x_cur = x_nxt; x_nxt = t; }
  }

  // ---- remaining atom-only layers ----
  for (int i = 3; i < 5; ++i) {
    conv(x_cur, x_nxt, N, srcA, dstA, eattr, E, cntAD, P_ALAYER0 + i * 10);
    { float* t = x_cur; x_cur = x_nxt; x_nxt = t; }
  }

  // ---- graph readout ----
  zero(bc, G);   count(batch, nullptr, N, bc);
  zero(lbc, G);  count(lmap, batch, E, lbc);      // batch[line_map[e]]
  zero(pooled, (size_t)G * 128);
  seg_sum(x_cur, batch, nullptr, N, pooled, 128, 0);
  seg_sum(l_cur, lmap, batch, E, pooled, 128, 64);
  hipLaunchKernelGGL(k_pool_div, dim3(16), dim3(256), 0, stream, pooled, bc, lbc);

  // out = silu(pooled @ Wout + bout)
  float* out = (float*)d_out;
  gemm(out, pooled, 128, nullptr, nullptr, 0, nullptr, P(P_OUT_W), P(P_OUT_B),
       nullptr, G, 128, 0);
  hipLaunchKernelGGL(k_silu, dim3((G * 64 + 255) / 256), dim3(256), 0, stream,
                     out, (size_t)G * 64);
}